// SConvLayer_40020505264736
// MI455X (gfx1250) — compile-verified
//
#include <hip/hip_runtime.h>
#include <math.h>
#include <stdint.h>

// ---------------------------------------------------------------------------
// PhazorSConv: y[b,t,d] = gate[d] * Re(w[b,t,d]),
//   w[b,-1,d] = last_conv[d],  w[b,t,d] = p[d]*w[b,t-1,d] + init[d]*x[b,t,d]
// Chunked parallel scan: C=64 chunks of S=64 steps along L=4096.
// ---------------------------------------------------------------------------

#if defined(__HIP_DEVICE_COMPILE__) && \
    __has_builtin(__builtin_amdgcn_global_load_async_to_lds_b128) && \
    __has_builtin(__builtin_amdgcn_s_wait_asynccnt)
#define USE_ASYNC 1
#else
#define USE_ASYNC 0
#endif

// Types matching the builtin's parameters as printed by clang's diagnostic:
//   param0: __attribute__((vector_size(16))) int __device__(as1) *
//   param1: LDS-side pointer (as3), same vector pointee
typedef int v4i __attribute__((vector_size(4 * sizeof(int))));
typedef __attribute__((address_space(1))) v4i as1_v4i;  // global
typedef __attribute__((address_space(3))) v4i as3_v4i;  // LDS

namespace {
constexpr int kB = 4;
constexpr int kL = 4096;
constexpr int kD = 1024;
constexpr int kC = 64;              // chunks along L
constexpr int kS = kL / kC;         // 64 steps per chunk
constexpr int kRows = 8;            // rows per async stage
constexpr int kStages = kS / kRows; // 8 stages per chunk
}

// Issue one async 16B global->LDS copy (per-lane addresses).
__device__ __forceinline__ void async_cp16(const float* g, float* l) {
#if USE_ASYNC
  __builtin_amdgcn_global_load_async_to_lds_b128(
      (as1_v4i*)(void*)g, (as3_v4i*)(void*)l, 0, 0);
#endif
}

// One complex scan step for one channel: w = p*w + init*x  (A = {p_re,p_im,i_re,i_im})
__device__ __forceinline__ void cstep(const float4& A, float xr, float& wre, float& wim) {
  float nr = fmaf(A.x, wre, fmaf(-A.y, wim, A.z * xr));
  float ni = fmaf(A.x, wim, fmaf(A.y, wre, A.w * xr));
  wre = nr;
  wim = ni;
}

// ---------------------------------------------------------------------------
// Kernel 0: per-channel constants (transcendentals hoisted out of hot loops)
//   cA[d] = (p_re, p_im, init_re, init_im)
//   cB[d] = (q_re, q_im, gate, 0) with q = p^S
// ---------------------------------------------------------------------------
__global__ void phazor_precompute(const float* __restrict__ angle,
                                  const float* __restrict__ lnmla,
                                  const float* __restrict__ init_re,
                                  const float* __restrict__ init_im,
                                  const float* __restrict__ scale,
                                  float4* __restrict__ cA,
                                  float4* __restrict__ cB) {
  int d = blockIdx.x * blockDim.x + threadIdx.x;
  if (d >= kD) return;
  float sc = scale[d];
  float decay = expf(lnmla[d]);
  float m = expf(-decay * sc);        // |p|
  float a = angle[d] * sc;            // arg(p)
  float pre = m * cosf(a);
  float pim = m * sinf(a);
  float gate = 1.0f - m;
  float qm = expf(-decay * sc * (float)kS);
  float qa = a * (float)kS;
  cA[d] = make_float4(pre, pim, init_re[d], init_im[d]);
  cB[d] = make_float4(qm * cosf(qa), qm * sinf(qa), gate, 0.0f);
}

// ---------------------------------------------------------------------------
// Kernels 1 & 3: chunk-local scan. Block = 256 threads = 1024 channels (4/thread),
// grid = B*C. Phase 1 (WRITE_Y=false): w starts at 0, store chunk-end states.
// Phase 3 (WRITE_Y=true): w starts at incoming prefix state, write gated output.
// x tile is staged global->LDS with async-copy double buffering when available.
// ---------------------------------------------------------------------------
template <bool WRITE_Y>
__global__ __launch_bounds__(256) void phazor_scan(const float* __restrict__ x,
                                                   const float4* __restrict__ cA,
                                                   const float4* __restrict__ cB,
                                                   const float2* __restrict__ ipre,
                                                   float2* __restrict__ wend,
                                                   float* __restrict__ y) {
  const int c = blockIdx.x & (kC - 1);
  const int b = blockIdx.x >> 6;
  const int d0 = threadIdx.x * 4;

  float4 A[4];
  float gt[4];
  float wre[4], wim[4];
#pragma unroll
  for (int k = 0; k < 4; ++k) {
    A[k] = cA[d0 + k];
    wre[k] = 0.0f;
    wim[k] = 0.0f;
    gt[k] = 0.0f;
  }
  if (WRITE_Y) {
    const float2* ip = ipre + (size_t)(b * kC + c) * kD + d0;
#pragma unroll
    for (int k = 0; k < 4; ++k) {
      float2 w0 = ip[k];
      wre[k] = w0.x;
      wim[k] = w0.y;
      gt[k] = cB[d0 + k].z;
    }
  }

  const size_t srow = (size_t)(b * kL + c * kS) * kD + d0;
  const float* xp = x + srow;
  float* yp = y + srow;

#if USE_ASYNC
  // 2 x 8 rows x 1024 ch x 4B = 64 KB LDS; each lane async-copies and later
  // reads back only its own 16B column slices -> no cross-thread barrier needed,
  // only per-wave s_wait_asynccnt ordering (async loads complete in order).
  __shared__ float lbuf[2][kRows][kD];
#pragma unroll
  for (int r = 0; r < kRows; ++r)
    async_cp16(xp + (size_t)r * kD, &lbuf[0][r][d0]);
  for (int st = 0; st < kStages; ++st) {
    if (st + 1 < kStages) {
#pragma unroll
      for (int r = 0; r < kRows; ++r)
        async_cp16(xp + (size_t)((st + 1) * kRows + r) * kD,
                   &lbuf[(st + 1) & 1][r][d0]);
      __builtin_amdgcn_s_wait_asynccnt(kRows);  // previous stage complete
    } else {
      __builtin_amdgcn_s_wait_asynccnt(0);
    }
    const float* lb = &lbuf[st & 1][0][0];
#pragma unroll
    for (int r = 0; r < kRows; ++r) {
      float4 xv = *(const float4*)(lb + (size_t)r * kD + d0);
      float xs[4] = {xv.x, xv.y, xv.z, xv.w};
#pragma unroll
      for (int k = 0; k < 4; ++k) cstep(A[k], xs[k], wre[k], wim[k]);
      if (WRITE_Y) {
        float4 yv = make_float4(gt[0] * wre[0], gt[1] * wre[1],
                                gt[2] * wre[2], gt[3] * wre[3]);
        *(float4*)(yp + (size_t)(st * kRows + r) * kD) = yv;
      }
    }
  }
#else
#pragma unroll 4
  for (int s = 0; s < kS; ++s) {
    if ((s & 7) == 0 && s + 8 < kS)
      __builtin_prefetch(xp + (size_t)(s + 8) * kD, 0, 0);  // global_prefetch_b8
    float4 xv = *(const float4*)(xp + (size_t)s * kD);
    float xs[4] = {xv.x, xv.y, xv.z, xv.w};
#pragma unroll
    for (int k = 0; k < 4; ++k) cstep(A[k], xs[k], wre[k], wim[k]);
    if (WRITE_Y) {
      float4 yv = make_float4(gt[0] * wre[0], gt[1] * wre[1],
                              gt[2] * wre[2], gt[3] * wre[3]);
      *(float4*)(yp + (size_t)s * kD) = yv;
    }
  }
#endif

  if (!WRITE_Y) {
    float2* op = wend + (size_t)(b * kC + c) * kD + d0;
#pragma unroll
    for (int k = 0; k < 4; ++k) op[k] = make_float2(wre[k], wim[k]);
  }
}

// ---------------------------------------------------------------------------
// Kernel 2: cross-chunk prefix scan. ipre[b,c,d] = state entering chunk c.
//   ipre[b,0,d] = last_conv[d];  ipre[b,c,d] = q*ipre[b,c-1,d] + wend[b,c-1,d]
// 4096 sequences x 64 steps; ~4 MiB traffic, negligible.
// ---------------------------------------------------------------------------
__global__ void phazor_chunk_prefix(const float2* __restrict__ wend,
                                    const float4* __restrict__ cB,
                                    const float* __restrict__ last_re,
                                    const float* __restrict__ last_im,
                                    float2* __restrict__ ipre) {
  int t = blockIdx.x * blockDim.x + threadIdx.x;
  if (t >= kB * kD) return;
  int d = t & (kD - 1);
  int b = t >> 10;
  float4 q4 = cB[d];
  float qre = q4.x, qim = q4.y;
  float ire = last_re[d], iim = last_im[d];
  size_t base = (size_t)b * kC * kD + d;
  ipre[base] = make_float2(ire, iim);
  for (int cc = 1; cc < kC; ++cc) {
    if (cc + 8 < kC) __builtin_prefetch(&wend[base + (size_t)(cc + 7) * kD], 0, 0);
    float2 e = wend[base + (size_t)(cc - 1) * kD];
    float nr = fmaf(qre, ire, fmaf(-qim, iim, e.x));
    float ni = fmaf(qre, iim, fmaf(qim, ire, e.y));
    ire = nr;
    iim = ni;
    ipre[base + (size_t)cc * kD] = make_float2(ire, iim);
  }
}

// ---------------------------------------------------------------------------
// Host-side launcher
// ---------------------------------------------------------------------------
extern "C" void kernel_launch(void* const* d_in, const int* in_sizes, int n_in,
                              void* d_out, int out_size, void* d_ws, size_t ws_size,
                              hipStream_t stream) {
  const float* x       = (const float*)d_in[0];
  const float* init_re = (const float*)d_in[1];
  const float* init_im = (const float*)d_in[2];
  const float* angle   = (const float*)d_in[3];
  const float* lnmla   = (const float*)d_in[4];
  const float* last_re = (const float*)d_in[5];
  const float* last_im = (const float*)d_in[6];
  const float* scale   = (const float*)d_in[7];

  char* ws = (char*)d_ws;
  float4* cA   = (float4*)ws;                         // kD * 16 B   = 16 KiB
  float4* cB   = (float4*)(ws + (size_t)kD * 16);     // kD * 16 B   = 16 KiB
  float2* wend = (float2*)(ws + (size_t)kD * 32);     // B*C*D * 8 B = 2 MiB
  float2* ipre = (float2*)(ws + (size_t)kD * 32 + (size_t)kB * kC * kD * 8);  // 2 MiB
  float* y = (float*)d_out;

  phazor_precompute<<<kD / 256, 256, 0, stream>>>(angle, lnmla, init_re, init_im,
                                                  scale, cA, cB);
  phazor_scan<false><<<kB * kC, 256, 0, stream>>>(x, cA, cB, ipre, wend, y);
  phazor_chunk_prefix<<<(kB * kD) / 256, 256, 0, stream>>>(wend, cB, last_re,
                                                           last_im, ipre);
  phazor_scan<true><<<kB * kC, 256, 0, stream>>>(x, cA, cB, ipre, wend, y);
}